// RoiAlign_18416819765907
// MI455X (gfx1250) — compile-verified
//
#include <hip/hip_runtime.h>

// 3D ROI-Align for MI455X (gfx1250, wave32).
// Memory-bound gather op (≈0.02 flop/byte): optimized for data movement.
//  - Two-stage LDS staging: 21 axis samples -> 343 fused per-cell descriptors
//  - Row-footprint global_prefetch_b8 warm pass (overlapped with descriptor build)
//  - Zero integer divides in the steady-state loop (carried counters)
//  - Coalesced non-temporal output stores (86 MB streaming)

namespace {
constexpr int RR = 512;          // rois
constexpr int BB = 2;            // batch
constexpr int CC = 128;          // channels
constexpr int TT = 64;
constexpr int WT  = 64 * 64;       // 4096   (H stride, elements)
constexpr int HWT = 64 * WT;       // 262144 (C stride, elements)
constexpr int PH = 7, PW = 7, PT = 7;
constexpr int CELLS = PH * PW * PT; // 343
constexpr int CPB = 32;            // channels per block
constexpr int CBLK = CC / CPB;     // 4 channel-blocks
constexpr int NT = 256;            // 8 waves (wave32)
constexpr int NROWPAIRS = 14 * 14; // (y-corner, x-corner) combos per ROI
constexpr float INV_IMG = 1.0f / 64.0f;
}

__global__ __launch_bounds__(NT) void roi_align3d_kernel(
    const float* __restrict__ feat,
    const float* __restrict__ rois,
    const int*   __restrict__ roi_idx,
    float*       __restrict__ out)
{
  // Stage-1: per-axis samples (y->H stride 4096, x->W stride 64, z->T stride 1)
  __shared__ int   a_o0[3][PH];
  __shared__ int   a_o1[3][PH];
  __shared__ float a_fr[3][PH];
  __shared__ float a_vd[3][PH];
  __shared__ int   s_base;
  // Stage-2: fused per-cell descriptors (SoA -> conflict-free lane-consecutive reads)
  __shared__ int   c_o00[CELLS], c_o01[CELLS], c_o10[CELLS], c_o11[CELLS];
  __shared__ int   c_z0[CELLS],  c_z1[CELLS];
  __shared__ float c_fz[CELLS],  c_fx[CELLS], c_gym[CELLS], c_fym[CELLS];

  const int r   = blockIdx.y;
  const int c0  = blockIdx.x * CPB;
  const int tid = threadIdx.x;

  // ---- Stage 1: axis sample descriptors (21 lanes) --------------------------
  if (tid < 3 * PH) {
    const int axis = tid / PH;          // 0:y  1:x  2:z
    const int j    = tid - axis * PH;   // sample index 0..6
    // boxes = rois/64, permute [0,1,3,4,2,5]: y<-cols(0,3) x<-(1,4) z<-(2,5)
    const float lo = rois[r * 6 + axis]     * INV_IMG;
    const float hi = rois[r * 6 + axis + 3] * INV_IMG;
    // match reference: scale = (hi-lo)*(dim-1)/(n-1); pos = lo*(dim-1)+j*scale
    const float scale = (hi - lo) * 63.0f / 6.0f;
    const float pos   = lo * 63.0f + (float)j * scale;
    const float fl = floorf(pos);
    const int i0 = (int)fminf(fmaxf(fl,        0.0f), 63.0f);
    const int i1 = (int)fminf(fmaxf(fl + 1.0f, 0.0f), 63.0f);
    const int mult = (axis == 0) ? WT : ((axis == 1) ? TT : 1);
    a_o0[axis][j] = i0 * mult;
    a_o1[axis][j] = i1 * mult;
    a_fr[axis][j] = pos - fl;
    a_vd[axis][j] = (pos >= 0.0f && pos <= 63.0f) ? 1.0f : 0.0f;
  }
  if (tid == 3 * PH) {
    int b = roi_idx[r];
    b = (b < 0) ? 0 : ((b >= BB) ? (BB - 1) : b);  // safety clamp
    s_base = b * (CC * HWT);
  }
  __syncthreads();

  const float* fbase = feat + s_base + c0 * HWT;

  // ---- Row-footprint prefetch (global_prefetch_b8, counter-free) ------------
  // The gather footprint is 14x14 (y,x) corner rows x CPB channels; each row's
  // z-span is <=~100B. Prefetch both span ends -> covers every line the compute
  // pass will touch, with no pz duplication. Issued *before* the stage-2 build
  // so the HBM pulls overlap with descriptor construction.
  {
    const int zA = a_o0[2][0];        // first z sample, lower corner
    const int zB = a_o1[2][PT - 1];   // last z sample, upper corner
    for (int n = tid; n < CPB * NROWPAIRS; n += NT) {
      const int c    = n / NROWPAIRS;
      const int pair = n - c * NROWPAIRS;
      const int yi   = pair / 14;
      const int xi   = pair - yi * 14;
      const int yo = (yi < PH) ? a_o0[0][yi] : a_o1[0][yi - PH];
      const int xo = (xi < PW) ? a_o0[1][xi] : a_o1[1][xi - PW];
      const float* row = fbase + c * HWT + yo + xo;
      __builtin_prefetch(row + zA, 0, 3);
      __builtin_prefetch(row + zB, 0, 3);
    }
  }

  // ---- Stage 2: fuse into per-cell descriptors (reused by all 32 channels) --
  for (int cell = tid; cell < CELLS; cell += NT) {
    const int py  = cell / (PW * PT);
    const int rem = cell - py * (PW * PT);
    const int px  = rem / PT;
    const int pz  = rem - px * PT;
    const int yo0 = a_o0[0][py], yo1 = a_o1[0][py];
    const int xo0 = a_o0[1][px], xo1 = a_o1[1][px];
    c_o00[cell] = yo0 + xo0;
    c_o01[cell] = yo0 + xo1;
    c_o10[cell] = yo1 + xo0;
    c_o11[cell] = yo1 + xo1;
    c_z0[cell]  = a_o0[2][pz];
    c_z1[cell]  = a_o1[2][pz];
    const float m = a_vd[0][py] * a_vd[1][px] * a_vd[2][pz]; // extrap==0 -> mask==mul
    const float fy = a_fr[0][py];
    c_fz[cell]  = a_fr[2][pz];
    c_fx[cell]  = a_fr[1][px];
    c_gym[cell] = (1.0f - fy) * m;
    c_fym[cell] = fy * m;
  }
  __syncthreads();

  float*    obase = out + (size_t)(r * CC + c0) * CELLS;
  const int nOut  = CPB * CELLS;  // 10976 outputs per block

  // ---- Main pass: gather + trilinear blend, coalesced NT stores -------------
  {
    int c = 0, cell = tid;             // tid < 256 < CELLS
    for (int k = tid; k < nOut; k += NT) {
      const float* fc = fbase + c * HWT;
      const int   o00 = c_o00[cell], o01 = c_o01[cell];
      const int   o10 = c_o10[cell], o11 = c_o11[cell];
      const int   z0  = c_z0[cell],  z1  = c_z1[cell];
      const float fz  = c_fz[cell],  fx  = c_fx[cell];
      const float gym = c_gym[cell], fym = c_fym[cell];

      // 8 independent gathers -> single wait, good MLP within the wave.
      const float v000 = fc[o00 + z0], v001 = fc[o00 + z1];
      const float v010 = fc[o01 + z0], v011 = fc[o01 + z1];
      const float v100 = fc[o10 + z0], v101 = fc[o10 + z1];
      const float v110 = fc[o11 + z0], v111 = fc[o11 + z1];

      const float gz = 1.0f - fz, gx = 1.0f - fx;
      const float z00 = v000 * gz + v001 * fz;
      const float z01 = v010 * gz + v011 * fz;
      const float z10 = v100 * gz + v101 * fz;
      const float z11 = v110 * gz + v111 * fz;
      const float res = (z00 * gx + z01 * fx) * gym + (z10 * gx + z11 * fx) * fym;

      __builtin_nontemporal_store(res, obase + k);
      cell += NT; if (cell >= CELLS) { cell -= CELLS; ++c; }
    }
  }
}

extern "C" void kernel_launch(void* const* d_in, const int* in_sizes, int n_in,
                              void* d_out, int out_size, void* d_ws, size_t ws_size,
                              hipStream_t stream) {
  (void)in_sizes; (void)n_in; (void)d_ws; (void)ws_size; (void)out_size;
  const float* feat = (const float*)d_in[0];   // (2,128,64,64,64) fp32
  const float* rois = (const float*)d_in[1];   // (512,6) fp32
  const int*   idx  = (const int*)d_in[2];     // (512,) int
  float*       out  = (float*)d_out;           // (512,128,7,7,7) fp32

  dim3 grid(CBLK, RR);   // (4, 512) blocks
  dim3 block(NT);        // 256 threads = 8 wave32 waves
  roi_align3d_kernel<<<grid, block, 0, stream>>>(feat, rois, idx, out);
}